// reduction_layer_86002425135161
// MI455X (gfx1250) — compile-verified
//
#include <hip/hip_runtime.h>
#include <math.h>

// MI455X / gfx1250 implementation.
// Problem sizes fixed by the reference: b=4, n=8192, e=64, NH=16, keep=4915.
#define N_PTS  8192
#define BATCH  4
#define EDIM   64
#define KNH    16
#define NKEEP  4915   // int((1.0 - 0.4) * 8192)

typedef __attribute__((ext_vector_type(2))) float v2f;
typedef __attribute__((ext_vector_type(4))) float f4;
typedef __attribute__((ext_vector_type(8))) float v8f;

// ---------------------------------------------------------------------------
// Kernel 1: fused KNN (WMMA Gram tiles) + local_dist.
// Block = 256 threads = 8 waves; each wave owns 32 query points (two 16-row
// A tiles sharing one 16-candidate B tile -> 2x V_WMMA_F32_16X16X4_F32 per
// candidate tile). All 32 lanes then run per-query top-16 selection: lane L
// owns query i0+L and reads its Gram row from the spilled D tile L>>4.
// K=2 real channels; K=2,3 zero-padded in lanes 16..31 of A and B operands.
// ---------------------------------------------------------------------------
__global__ __launch_bounds__(256)
void knn_localdist_kernel(const float* __restrict__ x,
                          const float* __restrict__ coords,
                          float* __restrict__ local_dist)
{
    // Per-wave spill of the two WMMA D tiles + per-candidate coord norms.
    __shared__ float spill[8][2][8][32];   // [wave][tile][vgpr r][lane]
    __shared__ float sqsp[8][16];          // [wave][candidate N]

    const int lane = threadIdx.x & 31;
    const int wave = threadIdx.x >> 5;
    const int bb   = blockIdx.y;
    const int i0   = (blockIdx.x * 8 + wave) * 32;   // first query of this wave

    const float* cx = coords + (size_t)(bb * 2 + 0) * N_PTS;
    const float* cy = coords + (size_t)(bb * 2 + 1) * N_PTS;

    // A operands (16x4, f32): K=0 -> VGPR0 lanes 0..15, K=1 -> VGPR1
    // lanes 0..15; lanes 16..31 carry K=2,3 and are zeroed (padding).
    float a0l = 0.0f, a1l = 0.0f, a0h = 0.0f, a1h = 0.0f;
    if (lane < 16) {
        a0l = cx[i0 + lane];        a1l = cy[i0 + lane];
        a0h = cx[i0 + 16 + lane];   a1h = cy[i0 + 16 + lane];
    }
    v2f Alo; Alo.x = a0l; Alo.y = a1l;
    v2f Ahi; Ahi.x = a0h; Ahi.y = a1h;

    // Every lane owns query i0+lane: its own coord norm.
    const float qx  = cx[i0 + lane];
    const float qy  = cy[i0 + lane];
    const float sqi = qx * qx + qy * qy;

    // Per-lane Gram-row location in the spill: tile, VGPR row, lane-half off.
    const int tile = lane >> 4;
    const int m    = lane & 15;
    const int row  = m & 7;
    const int off  = (m >> 3) << 4;

    // Top-16 selection state (registers only: constant-index unrolled access).
    float vals[KNH]; int idxs[KNH];
#pragma unroll
    for (int s = 0; s < KNH; ++s) { vals[s] = __builtin_inff(); idxs[s] = 0; }
    float vmax = __builtin_inff();
    int   slot = 0;

    for (int j0 = 0; j0 < N_PTS; j0 += 16) {
        // B operand (4x16, f32): same half-wave K split as A; zero-pad K=2,3.
        float b0 = 0.0f, b1 = 0.0f;
        if (lane < 16) {
            b0 = cx[j0 + lane];
            b1 = cy[j0 + lane];
            sqsp[wave][lane] = b0 * b0 + b1 * b1;
        }
        v2f B; B.x = b0; B.y = b1;

        v8f c0 = {0.f, 0.f, 0.f, 0.f, 0.f, 0.f, 0.f, 0.f};
        v8f c1 = {0.f, 0.f, 0.f, 0.f, 0.f, 0.f, 0.f, 0.f};
        // Two 16x16 Gram tiles against the same candidate tile.
        v8f accLo = __builtin_amdgcn_wmma_f32_16x16x4_f32(
            false, Alo, false, B, (short)0, c0, false, false);
        v8f accHi = __builtin_amdgcn_wmma_f32_16x16x4_f32(
            false, Ahi, false, B, (short)0, c1, false, false);

        // Spill D tiles; same-wave LDS ops are in-order, no barrier needed.
#pragma unroll
        for (int r = 0; r < 8; ++r) {
            spill[wave][0][r][lane] = accLo[r];
            spill[wave][1][r][lane] = accHi[r];
        }

        // All 32 lanes: scan the 16 candidates for their own query row.
#pragma unroll
        for (int cnd = 0; cnd < 16; ++cnd) {
            const float g  = spill[wave][tile][row][off + cnd];
            const float d2 = sqi + sqsp[wave][cnd] - 2.0f * g;
            if (d2 < vmax) {         // strict '<' == top_k low-index tie-break
                const int j = j0 + cnd;
#pragma unroll
                for (int s = 0; s < KNH; ++s) {
                    vals[s] = (s == slot) ? d2 : vals[s];
                    idxs[s] = (s == slot) ? j  : idxs[s];
                }
                vmax = vals[0]; slot = 0;
#pragma unroll
                for (int s = 1; s < KNH; ++s)
                    if (vals[s] > vmax) { vmax = vals[s]; slot = s; }
            }
        }
        // EXEC re-converges here: next WMMA sees all-1s EXEC.
    }

    // Phase 2 (all 32 lanes): local_dist via the norm identity
    //   sum_{k != m} ||x_k - x_m||^2 = S1 + 16*||x_m||^2 - 2 * x_m . S2
    {
        const int i = i0 + lane;

        // nearest entry (top-1 of top_k: smallest d2, ties -> smallest index)
        float bv = vals[0]; int bi = idxs[0];
#pragma unroll
        for (int s = 1; s < KNH; ++s) {
            const bool better =
                (vals[s] < bv) || ((vals[s] == bv) && (idxs[s] < bi));
            bv = better ? vals[s] : bv;
            bi = better ? idxs[s] : bi;
        }

        const float* xb = x + (size_t)bb * N_PTS * EDIM;
        const f4* xm = (const f4*)(xb + (size_t)bi * EDIM);
        f4 mreg[16];
#pragma unroll
        for (int e = 0; e < 16; ++e) mreg[e] = xm[e];

        float nm = 0.0f;
#pragma unroll
        for (int e = 0; e < 16; ++e)
            nm += mreg[e].x * mreg[e].x + mreg[e].y * mreg[e].y +
                  mreg[e].z * mreg[e].z + mreg[e].w * mreg[e].w;

        float S1 = 0.0f, dot = 0.0f;
        for (int k = 0; k < KNH; ++k) {
            const f4* rw = (const f4*)(xb + (size_t)idxs[k] * EDIM);
            float s1k = 0.0f, dk = 0.0f;
#pragma unroll
            for (int e = 0; e < 16; ++e) {
                const f4 rv = rw[e];
                s1k += rv.x * rv.x + rv.y * rv.y + rv.z * rv.z + rv.w * rv.w;
                dk  += mreg[e].x * rv.x + mreg[e].y * rv.y +
                       mreg[e].z * rv.z + mreg[e].w * rv.w;
            }
            S1 += s1k; dot += dk;
        }

        const float ld2 = S1 + 16.0f * nm - 2.0f * dot;
        local_dist[(size_t)bb * N_PTS + i] = sqrtf(fmaxf(ld2, 0.0f));
    }
}

// ---------------------------------------------------------------------------
// Kernel 2: exact stable-argsort rank + scatter-gather of kept rows.
// rank[i] = #{j : ld_j > ld_i} + #{j < i : ld_j == ld_i}  == position of i in
// stable argsort(-local_dist). Deterministic, atomic-free.
// ---------------------------------------------------------------------------
__global__ __launch_bounds__(256)
void rank_gather_kernel(const float* __restrict__ x,
                        const float* __restrict__ coords,
                        const float* __restrict__ local_dist,
                        float* __restrict__ out_x,
                        float* __restrict__ out_coords)
{
    __shared__ float ld_lds[N_PTS];   // 32 KB of the 320 KB WGP LDS

    const int bb  = blockIdx.y;
    const int tid = threadIdx.x;
    const float* ld = local_dist + (size_t)bb * N_PTS;

    for (int k = tid; k < N_PTS; k += 256) ld_lds[k] = ld[k];
    __syncthreads();

    const int   i  = blockIdx.x * 256 + tid;
    const float my = ld_lds[i];

    int rank = 0;
    for (int j = 0; j < N_PTS; ++j) {
        const float v = ld_lds[j];
        rank += (v > my) ? 1 : ((v == my && j < i) ? 1 : 0);
    }

    if (rank < NKEEP) {
        const f4* src = (const f4*)(x + ((size_t)bb * N_PTS + i) * EDIM);
        f4* dst = (f4*)(out_x + ((size_t)bb * NKEEP + rank) * EDIM);
#pragma unroll
        for (int e = 0; e < 16; ++e) dst[e] = src[e];
        out_coords[((size_t)bb * 2 + 0) * NKEEP + rank] =
            coords[((size_t)bb * 2 + 0) * N_PTS + i];
        out_coords[((size_t)bb * 2 + 1) * NKEEP + rank] =
            coords[((size_t)bb * 2 + 1) * N_PTS + i];
    }
}

// ---------------------------------------------------------------------------
extern "C" void kernel_launch(void* const* d_in, const int* in_sizes, int n_in,
                              void* d_out, int out_size, void* d_ws, size_t ws_size,
                              hipStream_t stream)
{
    const float* x      = (const float*)d_in[0];   // (4, 8192, 64) f32
    const float* coords = (const float*)d_in[1];   // (4, 2, 8192, 1) f32

    // Output layout: x_out | coords_out | local_dist (flat, return order).
    float* out        = (float*)d_out;
    float* out_x      = out;                                        // 4*4915*64
    float* out_coords = out_x + (size_t)BATCH * NKEEP * EDIM;       // 4*2*4915
    float* out_ld     = out_coords + (size_t)BATCH * 2 * NKEEP;     // 4*8192

    // Kernel 1: 8 waves/block * 32 queries/wave = 256 queries per block.
    dim3 g1(N_PTS / 256, BATCH);
    knn_localdist_kernel<<<g1, 256, 0, stream>>>(x, coords, out_ld);

    // Kernel 2: rank + gather (reads local_dist written above on same stream).
    dim3 g2(N_PTS / 256, BATCH);
    rank_gather_kernel<<<g2, 256, 0, stream>>>(x, coords, out_ld,
                                               out_x, out_coords);
}